// ImageAlignment_84026740179594
// MI455X (gfx1250) — compile-verified
//
#include <hip/hip_runtime.h>
#include <math.h>

// ---------------------------------------------------------------------------
// Constants for this problem: B=64 images, H=W=512, P=512.
// ---------------------------------------------------------------------------
#define HW   (512 * 512)
#define NB   64
#define PI_F 3.14159265358979323846f

typedef __attribute__((ext_vector_type(2))) float v2f;
typedef __attribute__((ext_vector_type(8))) float v8f;

// ---------------------------------------------------------------------------
// DFT matrix init: F[j,k] = exp(-2*pi*i*j*k/512), plus conjugate imag plane.
// (j*k mod 512) keeps the angle argument exact in fp32.
// ---------------------------------------------------------------------------
__global__ void initDFTKernel(float* __restrict__ Fr, float* __restrict__ Fi,
                              float* __restrict__ Fic) {
    int t = blockIdx.x * blockDim.x + threadIdx.x;
    if (t >= HW) return;
    int j = t >> 9, k = t & 511;
    int m = (j * k) & 511;
    float ang = -2.0f * PI_F * (float)m / 512.0f;
    float c = cosf(ang), s = sinf(ang);
    Fr[t] = c;
    Fi[t] = s;
    Fic[t] = -s;   // conj(F) imaginary plane, for the inverse DFT
}

// ---------------------------------------------------------------------------
// Complex GEMM, 512x512x512. Each wave computes a 16x64 strip (4 WMMA tiles),
// reusing the A fragments (and the single -Ai negation) across 4 N-tiles.
// One workgroup (8 waves) covers a 16x512 row band -> grid (32, 1, batch).
// C = A * B with planar real/imag storage; A always complex, B complex is a
// template parameter. Conjugation is handled by passing the precomputed -Fi
// plane (f32 WMMA has no A/B neg modifier; NEG is C-only for F32).
// Fragment layouts follow the CDNA5 ISA f32 WMMA tables (wave32):
//   A frag: A[m0 + lane%16][k0 + 2*(lane/16) + j]      j = 0,1
//   B frag: B[k0 + 2*(lane/16) + j][n0 + lane%16]
//   C:      C[m0 + v + 8*(lane/16)][n0 + lane%16]      v = 0..7
// ---------------------------------------------------------------------------
template <bool BC>
__global__ void cgemm512(const float* __restrict__ Ar, const float* __restrict__ Ai,
                         long aStride,
                         const float* __restrict__ Br, const float* __restrict__ Bi,
                         long bStride,
                         float* __restrict__ Cr, float* __restrict__ Ci, long cStride) {
    const int lane = threadIdx.x;           // 0..31 (wave32)
    const int l = lane & 15;
    const int h = lane >> 4;
    const int m0 = blockIdx.x * 16;
    const int n0 = threadIdx.y * 64;        // 8 waves x 64 cols = full 512 width
    const long zb = blockIdx.z;

    Ar += zb * aStride;  Ai += zb * aStride;
    Br += zb * bStride;
    if (BC) Bi += zb * bStride;
    Cr += zb * cStride;  Ci += zb * cStride;

    const v8f vzero = {0.f, 0.f, 0.f, 0.f, 0.f, 0.f, 0.f, 0.f};
    v8f cr[4], ci[4];
    #pragma unroll
    for (int t = 0; t < 4; ++t) { cr[t] = vzero; ci[t] = vzero; }

    const int arow = (m0 + l) * 512;
    for (int k0 = 0; k0 < 512; k0 += 4) {
        const int ka = k0 + 2 * h;
        // Prefetch the next A chunk (lowers to global_prefetch_b8).
        __builtin_prefetch(Ar + arow + ka + 64, 0, 0);
        // A fragments, shared across the 4 N-tiles (8B aligned -> v2f load).
        v2f ar = *(const v2f*)(Ar + arow + ka);
        v2f ai = *(const v2f*)(Ai + arow + ka);
        v2f nai = -ai;
        const float* b0r = Br + ka * 512 + n0 + l;
        const float* b1r = Br + (ka + 1) * 512 + n0 + l;
        const float* b0i = BC ? (Bi + ka * 512 + n0 + l) : nullptr;
        const float* b1i = BC ? (Bi + (ka + 1) * 512 + n0 + l) : nullptr;
        #pragma unroll
        for (int t = 0; t < 4; ++t) {
            v2f br;
            br.x = b0r[t * 16];
            br.y = b1r[t * 16];
            cr[t] = __builtin_amdgcn_wmma_f32_16x16x4_f32(false, ar, false, br, (short)0, cr[t], false, false);
            ci[t] = __builtin_amdgcn_wmma_f32_16x16x4_f32(false, ai, false, br, (short)0, ci[t], false, false);
            if (BC) {
                v2f bi;
                bi.x = b0i[t * 16];
                bi.y = b1i[t * 16];
                // Cr += (-Ai)*Bi ; Ci += Ar*Bi
                cr[t] = __builtin_amdgcn_wmma_f32_16x16x4_f32(false, nai, false, bi, (short)0, cr[t], false, false);
                ci[t] = __builtin_amdgcn_wmma_f32_16x16x4_f32(false, ar,  false, bi, (short)0, ci[t], false, false);
            }
        }
    }

    #pragma unroll
    for (int t = 0; t < 4; ++t) {
        #pragma unroll
        for (int v = 0; v < 8; ++v) {
            const int row = m0 + v + 8 * h;
            const int col = n0 + t * 16 + l;
            Cr[row * 512 + col] = cr[t][v];
            Ci[row * 512 + col] = ci[t][v];
        }
    }
}

// ---------------------------------------------------------------------------
// Border median via 4-pass radix select (exact order statistics).
// ---------------------------------------------------------------------------
__device__ __forceinline__ unsigned keyOf(float f) {
    unsigned u = __float_as_uint(f);
    return (u & 0x80000000u) ? ~u : (u | 0x80000000u);
}
__device__ __forceinline__ float valOf(unsigned k) {
    unsigned u = (k & 0x80000000u) ? (k & 0x7FFFFFFFu) : ~k;
    return __uint_as_float(u);
}
__device__ __forceinline__ int borderOffset(int i, int r) {
    const int rw = r * 512;
    if (i < rw) return i;                                            // top rows
    if (i < 2 * rw) { int j = i - rw; return (512 - r) * 512 + j; }  // bottom rows
    int j = i - 2 * rw;
    int row = j / (2 * r), c = j % (2 * r);
    int y = r + row;
    int x = (c < r) ? c : (512 - 2 * r + c);
    return y * 512 + x;
}

__device__ float radixSelect(const float* img, int r, int n, int k) {
    __shared__ unsigned hist[256];
    __shared__ unsigned sPrefix;
    __shared__ int sRank;
    unsigned prefix = 0;
    int rank = k;
    for (int pass = 0; pass < 4; ++pass) {
        const int shift = 24 - 8 * pass;
        for (int i = threadIdx.x; i < 256; i += blockDim.x) hist[i] = 0u;
        __syncthreads();
        for (int i = threadIdx.x; i < n; i += blockDim.x) {
            unsigned key = keyOf(img[borderOffset(i, r)]);
            bool match = (pass == 0) || ((key >> (shift + 8)) == (prefix >> (shift + 8)));
            if (match) atomicAdd(&hist[(key >> shift) & 255u], 1u);
        }
        __syncthreads();
        if (threadIdx.x == 0) {
            unsigned cum = 0;
            for (int d = 0; d < 256; ++d) {
                unsigned c = hist[d];
                if (cum + c > (unsigned)rank) {
                    sRank = rank - (int)cum;
                    sPrefix = prefix | ((unsigned)d << shift);
                    break;
                }
                cum += c;
            }
        }
        __syncthreads();
        prefix = sPrefix;
        rank = sRank;
        __syncthreads();
    }
    return valOf(prefix);
}

__global__ void borderMedianKernel(const float* __restrict__ src, long planeStride,
                                   int radius, float* __restrict__ out) {
    const float* img = src + (long)blockIdx.x * planeStride;
    const int r = radius;
    const int n = 2 * r * 512 + (512 - 2 * r) * 2 * r;   // even
    const int k = n / 2 - 1;
    float a = radixSelect(img, r, n, k);
    __syncthreads();
    float b = radixSelect(img, r, n, k + 1);
    if (threadIdx.x == 0) out[blockIdx.x] = 0.5f * (a + b);
}

// ---------------------------------------------------------------------------
// Apodization (aporad = int(512*0.12) = 61).
// ---------------------------------------------------------------------------
__device__ __forceinline__ float apo1d(int i) {
    const int ap = 61;
    const float nm1 = 2.0f * ap - 1.0f;   // 121
    if (i < ap) return 0.5f * (1.0f - cosf(2.0f * PI_F * (float)i / nm1));
    if (i >= 512 - ap) {
        int u = ap + (i - (512 - ap));
        return 0.5f * (1.0f - cosf(2.0f * PI_F * (float)u / nm1));
    }
    return 1.0f;
}

__global__ void apodizeKernel(const float* __restrict__ src, const float* __restrict__ bg,
                              float* __restrict__ out) {
    long t = (long)blockIdx.x * blockDim.x + threadIdx.x;
    int b = (int)(t >> 18);
    int p = (int)(t & (HW - 1));
    int y = p >> 9, x = p & 511;
    float apo = apo1d(y) * apo1d(x);
    out[t] = src[t] * apo + bg[b] * (1.0f - apo);
}

// ---------------------------------------------------------------------------
// |fftshift(F)| * highpass filter   (filter: 1 - cos^2(r), 1 outside pi/2)
// ---------------------------------------------------------------------------
__device__ __forceinline__ float lpfilt(int y, int x) {
    float yy = -0.5f * PI_F + (float)y * (PI_F / 511.0f);
    float xx = -0.5f * PI_F + (float)x * (PI_F / 511.0f);
    float r = sqrtf(yy * yy + xx * xx);
    if (r > 0.5f * PI_F) return 1.0f;
    float c = cosf(r);
    return 1.0f - c * c;
}

__global__ void magShiftFiltKernel(const float* __restrict__ re, const float* __restrict__ im,
                                   float* __restrict__ out) {
    long t = (long)blockIdx.x * blockDim.x + threadIdx.x;
    int b = (int)(t >> 18);
    int p = (int)(t & (HW - 1));
    int y = p >> 9, x = p & 511;
    int sy = (y + 256) & 511, sx = (x + 256) & 511;
    long q = ((long)b << 18) + (sy << 9) + sx;
    float a = re[q], c = im[q];
    out[t] = sqrtf(a * a + c * c) * lpfilt(y, x);
}

// ---------------------------------------------------------------------------
// Bilinear sampling with background value outside bounds.
// ---------------------------------------------------------------------------
__device__ float bilinearTap(const float* img, int yi, int xi, float bg) {
    bool valid = (yi >= 0) && (yi < 512) && (xi >= 0) && (xi < 512);
    int yc = min(max(yi, 0), 511), xc = min(max(xi, 0), 511);
    float v = img[yc * 512 + xc];
    return valid ? v : bg;
}
__device__ float bilinear(const float* img, float y, float x, float bg) {
    float y0f = floorf(y), x0f = floorf(x);
    float wy = y - y0f, wx = x - x0f;
    int y0 = (int)y0f, x0 = (int)x0f;
    return bilinearTap(img, y0, x0, bg) * (1.f - wy) * (1.f - wx)
         + bilinearTap(img, y0, x0 + 1, bg) * (1.f - wy) * wx
         + bilinearTap(img, y0 + 1, x0, bg) * wy * (1.f - wx)
         + bilinearTap(img, y0 + 1, x0 + 1, bg) * wy * wx;
}

// ---------------------------------------------------------------------------
// Log-polar resample of the (already fftshifted) spectrum magnitude.
// ---------------------------------------------------------------------------
__global__ void logpolarKernel(const float* __restrict__ d, const float* __restrict__ bg,
                               float* __restrict__ out) {
    long t = (long)blockIdx.x * blockDim.x + threadIdx.x;
    int b = (int)(t >> 18);
    int p = (int)(t & (HW - 1));
    int tt = p >> 9, rr = p & 511;
    const float lnb = logf(512.0f * 1.1f / 2.0f) / 512.0f;
    float th = -PI_F * (float)tt / 512.0f;
    float radius = expf((float)rr * lnb);
    float y = radius * sinf(th) + 256.0f;
    float x = radius * cosf(th) + 256.0f;     // ellipse = H/W = 1
    out[t] = bilinear(d + ((long)b << 18), y, x, bg[b]);
}

// ---------------------------------------------------------------------------
// max |f1| per image (for eps in the cross-power spectrum).
// ---------------------------------------------------------------------------
__global__ void maxAbsKernel(const float* __restrict__ re, const float* __restrict__ im,
                             float* __restrict__ out) {
    __shared__ float sm[256];
    const float* r = re + (long)blockIdx.x * HW;
    const float* i2 = im + (long)blockIdx.x * HW;
    float m = 0.0f;
    for (int i = threadIdx.x; i < HW; i += 256) {
        float a = r[i], b = i2[i];
        m = fmaxf(m, sqrtf(a * a + b * b));
    }
    sm[threadIdx.x] = m;
    __syncthreads();
    for (int s = 128; s > 0; s >>= 1) {
        if (threadIdx.x < s) sm[threadIdx.x] = fmaxf(sm[threadIdx.x], sm[threadIdx.x + s]);
        __syncthreads();
    }
    if (threadIdx.x == 0) out[blockIdx.x] = sm[0];
}

// ---------------------------------------------------------------------------
// r = f0 * conj(f1) / (|f0||f1| + eps), f0 broadcast (1 plane), f1 batched.
// ---------------------------------------------------------------------------
__global__ void crossPowerKernel(const float* __restrict__ f0r, const float* __restrict__ f0i,
                                 const float* __restrict__ f1r, const float* __restrict__ f1i,
                                 const float* __restrict__ mx,
                                 float* __restrict__ rr, float* __restrict__ ri) {
    long t = (long)blockIdx.x * blockDim.x + threadIdx.x;
    int b = (int)(t >> 18);
    long p = t & (long)(HW - 1);
    float ar = f0r[p], ai = f0i[p];
    float br = f1r[t], bi = f1i[t];
    float nr = ar * br + ai * bi;
    float ni = ai * br - ar * bi;
    float den = sqrtf(ar * ar + ai * ai) * sqrtf(br * br + bi * bi) + mx[b] * 1e-15f;
    rr[t] = nr / den;
    ri[t] = ni / den;
}

// ---------------------------------------------------------------------------
// scps = fftshift(|ifft2(r)|); 1/(512*512) ifft normalization folded in.
// ---------------------------------------------------------------------------
__global__ void scpsKernel(const float* __restrict__ yr, const float* __restrict__ yi,
                           float* __restrict__ out) {
    long t = (long)blockIdx.x * blockDim.x + threadIdx.x;
    int b = (int)(t >> 18);
    int p = (int)(t & (HW - 1));
    int y = p >> 9, x = p & 511;
    int sy = (y + 256) & 511, sx = (x + 256) & 511;
    long q = ((long)b << 18) + (sy << 9) + sx;
    float a = yr[q], c = yi[q];
    out[t] = sqrtf(a * a + c * c) * (1.0f / 262144.0f);
}

// ---------------------------------------------------------------------------
// argmax per image (first index wins ties), then 5x5 wrapped COM refinement.
// ---------------------------------------------------------------------------
__global__ void argmaxKernel(const float* __restrict__ arr, int* __restrict__ idx) {
    __shared__ float sv[256];
    __shared__ int si[256];
    const float* a = arr + (long)blockIdx.x * HW;
    float bv = -1e30f;
    int bi = 0;
    for (int i = threadIdx.x; i < HW; i += 256) {
        float v = a[i];
        if (v > bv) { bv = v; bi = i; }
    }
    sv[threadIdx.x] = bv;
    si[threadIdx.x] = bi;
    __syncthreads();
    for (int s = 128; s > 0; s >>= 1) {
        if (threadIdx.x < s) {
            if (sv[threadIdx.x + s] > sv[threadIdx.x] ||
                (sv[threadIdx.x + s] == sv[threadIdx.x] && si[threadIdx.x + s] < si[threadIdx.x])) {
                sv[threadIdx.x] = sv[threadIdx.x + s];
                si[threadIdx.x] = si[threadIdx.x + s];
            }
        }
        __syncthreads();
    }
    if (threadIdx.x == 0) {
        idx[2 * blockIdx.x] = si[0] >> 9;
        idx[2 * blockIdx.x + 1] = si[0] & 511;
    }
}

__device__ __forceinline__ float wrapCoord(float p) {
    float m = fmodf(p + 0.5f, 512.0f);
    if (m < 0.0f) m += 512.0f;
    return m - 0.5f;
}

__global__ void peakInterpKernel(const float* __restrict__ arr, const int* __restrict__ idx,
                                 float* __restrict__ peak) {
    if (threadIdx.x != 0) return;
    const int b = blockIdx.x;
    const float* a = arr + (long)b * HW;
    int iy = idx[2 * b], ix = idx[2 * b + 1];
    float s = 0.0f, s0 = 0.0f, s1 = 0.0f;
    for (int d0 = 0; d0 < 5; ++d0) {
        int y = (iy - 2 + d0 + 512) & 511;
        for (int d1 = 0; d1 < 5; ++d1) {
            int x = (ix - 2 + d1 + 512) & 511;
            float v = a[y * 512 + x];
            s += v;
            s0 += v * (float)d0;
            s1 += v * (float)d1;
        }
    }
    float p0 = (float)iy + s0 / s - 2.0f;
    float p1 = (float)ix + s1 / s - 2.0f;
    peak[2 * b] = wrapCoord(p0);
    peak[2 * b + 1] = wrapCoord(p1);
}

// ---------------------------------------------------------------------------
// scale / angle from the rotation-scale peak; tvec from the translation peak.
// ---------------------------------------------------------------------------
__global__ void angScaleKernel(const float* __restrict__ peak,
                               float* __restrict__ scaleA, float* __restrict__ angleA) {
    int b = threadIdx.x;
    if (b >= NB) return;
    float a0 = peak[2 * b] - 256.0f;
    float a1 = peak[2 * b + 1] - 256.0f;
    float ang = -180.0f * a0 / 512.0f;                 // degrees(-pi*arg0/P)
    float w = fmodf(ang + 180.0f, 360.0f);
    if (w < 0.0f) w += 360.0f;
    w -= 180.0f;
    angleA[b] = -w;
    const float lnb = logf(512.0f * 1.1f / 2.0f) / 512.0f;
    scaleA[b] = expf(-a1 * lnb);                        // 1 / log_base^arg1
}

__global__ void tvecKernel(const float* __restrict__ peak, float* __restrict__ tv) {
    int b = threadIdx.x;
    if (b >= NB) return;
    tv[2 * b] = peak[2 * b] - 256.0f;
    tv[2 * b + 1] = peak[2 * b + 1] - 256.0f;
}

// ---------------------------------------------------------------------------
// Inverse-mapped similarity transform + bilinear sampling.
// ---------------------------------------------------------------------------
__global__ void transformKernel(const float* __restrict__ img,
                                const float* __restrict__ scale, const float* __restrict__ angle,
                                const float* __restrict__ tvec, const float* __restrict__ bg,
                                float* __restrict__ out) {
    long t = (long)blockIdx.x * blockDim.x + threadIdx.x;
    int b = (int)(t >> 18);
    int p = (int)(t & (HW - 1));
    int y = p >> 9, x = p & 511;
    float s = scale[b];
    float a = angle[b] * (PI_F / 180.0f);
    float t0 = tvec ? tvec[2 * b] : 0.0f;
    float t1 = tvec ? tvec[2 * b + 1] : 0.0f;
    float ca = cosf(a), sa = sinf(a);
    float dy = (float)y - t0 - 255.5f;
    float dx = (float)x - t1 - 255.5f;
    float sy = (ca * dy - sa * dx) / s + 255.5f;
    float sx = (sa * dy + ca * dx) / s + 255.5f;
    out[t] = bilinear(img + ((long)b << 18), sy, sx, bg[b]);
}

// ---------------------------------------------------------------------------
// Host-side orchestration.
// ---------------------------------------------------------------------------
extern "C" void kernel_launch(void* const* d_in, const int* in_sizes, int n_in,
                              void* d_out, int out_size, void* d_ws, size_t ws_size,
                              hipStream_t stream) {
    (void)in_sizes; (void)n_in; (void)out_size; (void)ws_size;

    const float* images = (const float*)d_in[0];   // (64,512,512)
    const float* ref    = (const float*)d_in[1];   // (512,512)
    float* out          = (float*)d_out;

    float* W = (float*)d_ws;
    long off = 0;
    float* Fr  = W + off; off += HW;
    float* Fi  = W + off; off += HW;
    float* Fic = W + off; off += HW;    // -Fi (conjugate plane for inverse DFT)

    // scalar block
    float* SC    = W + off; off += 1024;
    float* bg30  = SC + 0;     // 64
    float* bg30r = SC + 64;    // 1
    float* bg5   = SC + 80;    // 64
    float* bgd1  = SC + 160;   // 64
    float* bgd0  = SC + 224;   // 1
    float* mx    = SC + 240;   // 64
    float* peak  = SC + 320;   // 128
    float* scaleA= SC + 448;   // 64
    float* angleA= SC + 512;   // 64
    float* tvA   = SC + 576;   // 128
    int*   idx   = (int*)(SC + 704); // 128 ints

    // ref-sized scratch planes (1 image each)
    float* R0 = W + off; off += HW;
    float* R1 = W + off; off += HW;
    float* R2 = W + off; off += HW;
    float* R3 = W + off; off += HW;
    float* R4 = W + off; off += HW;
    float* R5 = W + off; off += HW;

    // batch-sized planes (64 images each)
    const long BP = (long)NB * HW;
    float* buf0 = W + off; off += BP;
    float* buf1 = W + off; off += BP;
    float* buf2 = W + off; off += BP;
    float* buf3 = W + off; off += BP;
    float* buf4 = W + off; off += BP;
    float* buf5 = W + off; off += BP;
    float* buf6 = W + off; off += BP;

    const dim3 gBlk(32, 8, 1);                // 8 waves; each wave: 16x64 strip
    const dim3 gGridB(32, 1, NB);             // full batch (workgroup = 16x512 band)
    const dim3 gGrid1(32, 1, 1);              // single image
    const int EB = 256;
    const int nElemB = (int)(BP / EB);        // batch elementwise blocks
    const int nElem1 = HW / EB;               // single-plane elementwise blocks

    // ---- DFT matrices ----
    initDFTKernel<<<nElem1, EB, 0, stream>>>(Fr, Fi, Fic);

    // ---- border medians ----
    borderMedianKernel<<<NB, 256, 0, stream>>>(images, HW, 30, bg30);
    borderMedianKernel<<<1, 256, 0, stream>>>(ref, 0, 30, bg30r);
    borderMedianKernel<<<NB, 256, 0, stream>>>(images, HW, 5, bg5);

    // ---- apodize ----
    apodizeKernel<<<nElemB, EB, 0, stream>>>(images, bg30, buf0);
    apodizeKernel<<<nElem1, EB, 0, stream>>>(ref, bg30r, R0);

    // ---- fft2(apodized images): T = F*X ; Y = T*F ----
    cgemm512<false><<<gGridB, gBlk, 0, stream>>>(Fr, Fi, 0, buf0, nullptr, HW, buf1, buf2, HW);
    cgemm512<true ><<<gGridB, gBlk, 0, stream>>>(buf1, buf2, HW, Fr, Fi, 0, buf3, buf4, HW);
    magShiftFiltKernel<<<nElemB, EB, 0, stream>>>(buf3, buf4, buf0);      // d1 -> buf0

    // ---- fft2(apodized ref) ----
    cgemm512<false><<<gGrid1, gBlk, 0, stream>>>(Fr, Fi, 0, R0, nullptr, HW, R1, R2, HW);
    cgemm512<true ><<<gGrid1, gBlk, 0, stream>>>(R1, R2, HW, Fr, Fi, 0, R3, R4, HW);
    magShiftFiltKernel<<<nElem1, EB, 0, stream>>>(R3, R4, R5);            // d0 -> R5

    // ---- log-polar bg + resample ----
    borderMedianKernel<<<NB, 256, 0, stream>>>(buf0, HW, 5, bgd1);
    borderMedianKernel<<<1, 256, 0, stream>>>(R5, 0, 5, bgd0);
    logpolarKernel<<<nElemB, EB, 0, stream>>>(buf0, bgd1, buf1);          // lp1 -> buf1
    logpolarKernel<<<nElem1, EB, 0, stream>>>(R5, bgd0, R0);              // lp0 -> R0

    // ---- fft2(lp1), fft2(lp0) ----
    cgemm512<false><<<gGridB, gBlk, 0, stream>>>(Fr, Fi, 0, buf1, nullptr, HW, buf2, buf3, HW);
    cgemm512<true ><<<gGridB, gBlk, 0, stream>>>(buf2, buf3, HW, Fr, Fi, 0, buf4, buf5, HW); // f1
    cgemm512<false><<<gGrid1, gBlk, 0, stream>>>(Fr, Fi, 0, R0, nullptr, HW, R1, R2, HW);
    cgemm512<true ><<<gGrid1, gBlk, 0, stream>>>(R1, R2, HW, Fr, Fi, 0, R3, R4, HW);         // f0

    // ---- cross-power spectrum, ifft2 (conj planes), scps ----
    maxAbsKernel<<<NB, 256, 0, stream>>>(buf4, buf5, mx);
    crossPowerKernel<<<nElemB, EB, 0, stream>>>(R3, R4, buf4, buf5, mx, buf2, buf3);
    cgemm512<true ><<<gGridB, gBlk, 0, stream>>>(Fr, Fic, 0, buf2, buf3, HW, buf4, buf5, HW);
    cgemm512<true ><<<gGridB, gBlk, 0, stream>>>(buf4, buf5, HW, Fr, Fic, 0, buf0, buf1, HW);
    scpsKernel<<<nElemB, EB, 0, stream>>>(buf0, buf1, buf2);              // scps -> buf2

    // ---- peak -> scale/angle ----
    argmaxKernel<<<NB, 256, 0, stream>>>(buf2, idx);
    peakInterpKernel<<<NB, 32, 0, stream>>>(buf2, idx, peak);
    angScaleKernel<<<1, 64, 0, stream>>>(peak, scaleA, angleA);

    // ---- im_rs = transform(images, scale, angle, tvec=0) ----
    transformKernel<<<nElemB, EB, 0, stream>>>(images, scaleA, angleA, nullptr, bg5, buf0);

    // ---- translation: phase_correlation(ref, im_rs) ----
    cgemm512<false><<<gGrid1, gBlk, 0, stream>>>(Fr, Fi, 0, ref, nullptr, HW, R1, R2, HW);
    cgemm512<true ><<<gGrid1, gBlk, 0, stream>>>(R1, R2, HW, Fr, Fi, 0, R3, R4, HW);         // f0
    cgemm512<false><<<gGridB, gBlk, 0, stream>>>(Fr, Fi, 0, buf0, nullptr, HW, buf1, buf2, HW);
    cgemm512<true ><<<gGridB, gBlk, 0, stream>>>(buf1, buf2, HW, Fr, Fi, 0, buf3, buf4, HW); // f1
    maxAbsKernel<<<NB, 256, 0, stream>>>(buf3, buf4, mx);
    crossPowerKernel<<<nElemB, EB, 0, stream>>>(R3, R4, buf3, buf4, mx, buf1, buf2);
    cgemm512<true ><<<gGridB, gBlk, 0, stream>>>(Fr, Fic, 0, buf1, buf2, HW, buf5, buf6, HW);
    cgemm512<true ><<<gGridB, gBlk, 0, stream>>>(buf5, buf6, HW, Fr, Fic, 0, buf3, buf4, HW);
    scpsKernel<<<nElemB, EB, 0, stream>>>(buf3, buf4, buf0);

    argmaxKernel<<<NB, 256, 0, stream>>>(buf0, idx);
    peakInterpKernel<<<NB, 32, 0, stream>>>(buf0, idx, peak);
    tvecKernel<<<1, 64, 0, stream>>>(peak, tvA);

    // ---- final transform -> d_out ----
    transformKernel<<<nElemB, EB, 0, stream>>>(images, scaleA, angleA, tvA, bg5, out);
}